// LSS_Core_30107720745185
// MI455X (gfx1250) — compile-verified
//
#include <hip/hip_runtime.h>

// ---- problem constants (from reference) ----
#define CIN    512
#define D_Q    41
#define C_Q    64
#define OUTC   105      // D + C
#define HW     704      // H*W (704 = 11*64 -> 64-pixel tiles never cross an image)
#define NPIX   16896    // B*N*H*W
#define NPTS   692736   // B*N*D*H*W
#define NXG    200
#define NYG    200
#define XB0    (-50.0f)
#define XSTEP_INV 2.0f  // 1/0.5 exact in fp32

#define MT     64       // pixels per workgroup (M tile)
#define KC     32       // K chunk staged in LDS per double-buffer step

typedef float v2f __attribute__((ext_vector_type(2)));
typedef float v8f __attribute__((ext_vector_type(8)));
typedef int   v4i_vs __attribute__((__vector_size__(16)));

#define AS(n) __attribute__((address_space(n)))

// ---- CDNA5 async global->LDS copies (ASYNCcnt path), with safe fallback ----
// Probed prototypes (from clang diagnostics):
//   b32 : (AS1 int* src, AS3 int* dst, imm int offset, imm int cpol)
//   b128: (AS1 v4i* src, AS3 v4i* dst, imm int offset, imm int cpol)
__device__ __forceinline__ void g2l_b32(const float* g, float* l) {
#if __has_builtin(__builtin_amdgcn_global_load_async_to_lds_b32)
  __builtin_amdgcn_global_load_async_to_lds_b32((AS(1) int*)g, (AS(3) int*)l,
                                                0, 0);
#else
  *l = *g;
#endif
}

__device__ __forceinline__ void g2l_b128(const float* g, float* l) {
#if __has_builtin(__builtin_amdgcn_global_load_async_to_lds_b128)
  __builtin_amdgcn_global_load_async_to_lds_b128((AS(1) v4i_vs*)g,
                                                 (AS(3) v4i_vs*)l, 0, 0);
#else
  *(float4*)l = *(const float4*)g;
#endif
}

__device__ __forceinline__ void wait_async0() {
#if __has_builtin(__builtin_amdgcn_s_wait_asynccnt)
  __builtin_amdgcn_s_wait_asynccnt(0);
#elif __has_builtin(__builtin_amdgcn_global_load_async_to_lds_b32)
  asm volatile("s_wait_asynccnt 0x0" ::: "memory");
#endif
}

// LDS plan (floats), all aliased inside one 11392-float (44.5 KB) block:
//   [0      .. 4095 ]  A double buffer: [buf][sub(4)][cc(32)][m(16)]
//   [4096   .. 11263]  B double buffer: [buf][wave(7)][cc(32)][n(16)]
//   after K-loop, [0 .. 7167] is reused as feat[pix(64)][o(112)],
//   [7168..7231] softmax max, [7232..7295] softmax 1/sum.
#define A_OFF(buf) ((buf) * (4 * KC * 16))
#define B_OFF(buf) (4096 + (buf) * (7 * KC * 16))
#define SMEM_FLOATS 11392

// Kernel 1: fused 1x1-conv GEMM (V_WMMA_F32_16X16X4_F32) + depth softmax.
// Workgroup = 64-pixel M tile; 7 waves = 7 N tiles over 105 (pad 112) outputs;
// each wave holds 4 accumulators (4 M sub-tiles) sharing one B fragment.
// K=512 swept in 32-deep double-buffered LDS chunks fed by async copies.
__global__ __launch_bounds__(224) void lss_conv_softmax(
    const float* __restrict__ x, const float* __restrict__ conv_w,
    const float* __restrict__ conv_b, float* __restrict__ depth_ws,
    float* __restrict__ ctx_ws) {
  __shared__ float smem[SMEM_FLOATS];

  const int tid  = threadIdx.x;
  const int wave = tid >> 5;             // 0..6 (wave32)
  const int lane = tid & 31;
  const int m    = lane & 15;            // M row (A) / N col (B) for this lane
  const int koff = (lane >> 4) << 1;     // lanes 0-15 -> K 0,1 ; 16-31 -> K 2,3

  const int p0  = blockIdx.x * MT;       // first pixel of tile
  const int bn  = p0 / HW;               // image index; tile stays inside image
  const int hw0 = p0 % HW;               // multiple of 64 -> 16B-aligned rows

  const int o = wave * 16 + m;           // output channel column for this lane
  const float* xbase = x + (size_t)bn * CIN * HW + hw0;

  // stage chunk k0 (KC deep) into buffer `buf`
  auto stage = [&](int k0, int buf) {
    // A: 64 pixels x KC channels, 16B async transfers (contiguous pixels)
    for (int g = tid; g < KC * 16; g += 224) {      // 512 b128 groups
      int cc = g >> 4, pg = g & 15;                 // pg = pixel/4 group
      int sub = pg >> 2, mm = (pg & 3) << 2;
      g2l_b128(xbase + (size_t)(k0 + cc) * HW + (pg << 2),
               &smem[A_OFF(buf) + sub * (KC * 16) + cc * 16 + mm]);
    }
    // B: conv_w[o, k0..k0+KC) transposed to [k][n], zero-pad o>=105
    for (int idx = tid; idx < 7 * KC * 16; idx += 224) {
      int w = idx / (KC * 16), r = idx % (KC * 16);
      int cc = r >> 4, n = r & 15;
      int oo = w * 16 + n;
      if (oo < OUTC)
        g2l_b32(&conv_w[oo * CIN + k0 + cc], &smem[B_OFF(buf) + idx]);
      else
        smem[B_OFF(buf) + idx] = 0.0f;
    }
  };

  v8f acc0 = {}, acc1 = {}, acc2 = {}, acc3 = {};
  stage(0, 0);
  for (int c = 0; c < CIN / KC; ++c) {
    wait_async0();
    __syncthreads();                    // chunk c visible; buf c^1 free to restage
    if (c + 1 < CIN / KC) stage(c + 1, (c + 1) & 1);
    const int ab = A_OFF(c & 1), bb = B_OFF(c & 1) + wave * (KC * 16);
#pragma unroll 4
    for (int k = 0; k < KC; k += 4) {
      v2f b;
      b.x = smem[bb + (k + koff) * 16 + m];
      b.y = smem[bb + (k + koff + 1) * 16 + m];
      v2f a;
      a.x = smem[ab + (k + koff) * 16 + m];
      a.y = smem[ab + (k + koff + 1) * 16 + m];
      acc0 = __builtin_amdgcn_wmma_f32_16x16x4_f32(false, a, false, b, (short)0,
                                                   acc0, false, false);
      a.x = smem[ab + 512 + (k + koff) * 16 + m];
      a.y = smem[ab + 512 + (k + koff + 1) * 16 + m];
      acc1 = __builtin_amdgcn_wmma_f32_16x16x4_f32(false, a, false, b, (short)0,
                                                   acc1, false, false);
      a.x = smem[ab + 1024 + (k + koff) * 16 + m];
      a.y = smem[ab + 1024 + (k + koff + 1) * 16 + m];
      acc2 = __builtin_amdgcn_wmma_f32_16x16x4_f32(false, a, false, b, (short)0,
                                                   acc2, false, false);
      a.x = smem[ab + 1536 + (k + koff) * 16 + m];
      a.y = smem[ab + 1536 + (k + koff + 1) * 16 + m];
      acc3 = __builtin_amdgcn_wmma_f32_16x16x4_f32(false, a, false, b, (short)0,
                                                   acc3, false, false);
    }
    __syncthreads();                    // done reading buf before next restage
  }

  // reuse smem[0..7167] as feat[pix][112] (+ softmax stats above it)
  float* feat  = smem;
  float* s_max = smem + 7168;
  float* s_inv = smem + 7232;
  const int rbase = (lane < 16) ? 0 : 8;   // D layout: VGPR r -> M=r / M=r+8
  if (o < OUTC) {
    float cb = conv_b[o];
#pragma unroll
    for (int r = 0; r < 8; ++r) {
      feat[(0 * 16 + rbase + r) * 112 + o] = acc0[r] + cb;
      feat[(1 * 16 + rbase + r) * 112 + o] = acc1[r] + cb;
      feat[(2 * 16 + rbase + r) * 112 + o] = acc2[r] + cb;
      feat[(3 * 16 + rbase + r) * 112 + o] = acc3[r] + cb;
    }
  }
  __syncthreads();

  // per-pixel softmax stats over depth channels 0..40
  if (tid < MT) {
    float mx = -1e30f;
    for (int d = 0; d < D_Q; ++d) mx = fmaxf(mx, feat[tid * 112 + d]);
    float s = 0.0f;
    for (int d = 0; d < D_Q; ++d) s += __expf(feat[tid * 112 + d] - mx);
    s_max[tid] = mx;
    s_inv[tid] = 1.0f / s;
  }
  __syncthreads();

  // depth_ws: flat (B,N,D,H,W) layout -> matches coords/point flattening
  for (int idx = tid; idx < MT * D_Q; idx += 224) {
    int p = idx / D_Q, d = idx - p * D_Q;
    depth_ws[(bn * D_Q + d) * HW + hw0 + p] =
        __expf(feat[p * 112 + d] - s_max[p]) * s_inv[p];
  }
  // ctx_ws: [pixel][c]
  for (int idx = tid; idx < MT * C_Q; idx += 224) {
    int p = idx >> 6, cc = idx & 63;
    ctx_ws[(p0 + p) * C_Q + cc] = feat[p * 112 + D_Q + cc];
  }
}

// Kernel 2: voxel-pool scatter. 64 lanes per point (one per channel);
// float->int cast truncates toward zero, matching jnp astype(int32).
// Early-out on the voxel test before touching depth/ctx (~80% of points die).
__global__ __launch_bounds__(256) void lss_splat(
    const float* __restrict__ coords, const float* __restrict__ depth_ws,
    const float* __restrict__ ctx_ws, float* __restrict__ out) {
  int q = blockIdx.x * 4 + (threadIdx.x >> 6);
  if (q >= NPTS) return;
  int c = threadIdx.x & 63;
  float cx = coords[q * 3 + 0];
  float cy = coords[q * 3 + 1];
  float cz = coords[q * 3 + 2];
  int xi = (int)((cx - XB0) * XSTEP_INV);
  int yi = (int)((cy - XB0) * XSTEP_INV);
  int zi = (int)((cz - XB0) * XSTEP_INV);
  if (xi < 0 || xi >= NXG || yi < 0 || yi >= NYG || zi != 0) return;
  int hw = q % HW;
  int t  = q / HW;
  int bn = t / D_Q;
  float v = depth_ws[q] * ctx_ws[(bn * HW + hw) * C_Q + c];
  // out layout is (1, C, NX, NY): channel-major
  atomicAdd(&out[c * (NXG * NYG) + xi * NYG + yi], v);
}

__global__ void lss_zero(float* __restrict__ p, int n) {
  int i = blockIdx.x * blockDim.x + threadIdx.x;
  if (i < n) p[i] = 0.0f;
}

extern "C" void kernel_launch(void* const* d_in, const int* in_sizes, int n_in,
                              void* d_out, int out_size, void* d_ws, size_t ws_size,
                              hipStream_t stream) {
  const float* x      = (const float*)d_in[0];  // (B,N,512,16,44)
  const float* coords = (const float*)d_in[1];  // (B,N,41,16,44,3)
  const float* conv_w = (const float*)d_in[2];  // (105,512)
  const float* conv_b = (const float*)d_in[3];  // (105,)
  float* out = (float*)d_out;                   // (1,64,200,200)

  // workspace: depth (NPTS floats) + ctx (NPIX*64 floats) = ~7.1 MB
  float* depth_ws = (float*)d_ws;
  float* ctx_ws   = depth_ws + NPTS;

  lss_zero<<<(out_size + 255) / 256, 256, 0, stream>>>(out, out_size);
  lss_conv_softmax<<<NPIX / MT, 224, 0, stream>>>(x, conv_w, conv_b,
                                                  depth_ws, ctx_ws);
  lss_splat<<<(NPTS + 3) / 4, 256, 0, stream>>>(coords, depth_ws, ctx_ws, out);
}